// SparseArch_23484881174936
// MI455X (gfx1250) — compile-verified
//
#include <hip/hip_runtime.h>
#include <hip/hip_bf16.h>

// TBE pooled-embedding forward for MI455X (gfx1250, wave32) using the
// CDNA5 Tensor Data Mover in GATHER MODE: one tensor_load_to_lds pulls up to
// 8 embedding rows (32-bit row indices) into LDS; a bag of 20 rows = 3 TDM ops.
// The wave then reduces its private LDS tile and stores the pooled result.
//
// Toolchain note: this lane's clang-23 builtin takes 6 args; the 5th (int32x8)
// corresponds to the VIMAGE VADDR4 slot which the ISA defines as unused/NULL,
// so it is passed as zeros.

typedef unsigned int       u32;
typedef unsigned long long u64;
typedef u32 v4u __attribute__((ext_vector_type(4)));
typedef int v8i __attribute__((ext_vector_type(8)));
typedef int v4i __attribute__((ext_vector_type(4)));

#define T_TABLES 16
#define E_ROWS   200000
#define D_DIM    64
#define B_BATCH  4096
#define BAG_L    20
#define WAVES_PER_BLOCK 8                    // 256 threads = 8 wave32
#define ROW_BYTES (D_DIM * 4)                // 256 B per embedding row
#define BAG_BYTES (BAG_L * ROW_BYTES)        // 5120 B LDS slice per wave

// Issue one TDM gather: n_idx rows (32-bit indices) of a [E_ROWS x D_DIM] f32
// tensor at gaddr, written contiguously (n_idx x 256B) to LDS byte addr.
__device__ __forceinline__ void tdm_gather_rows(
    u64 gaddr, u32 lds_addr, int n_idx,
    int i0, int i1, int i2, int i3, int i4, int i5, int i6, int i7)
{
    v4u g0;
    g0[0] = 1u | (1u << 30) | (1u << 31);            // count=1, idx32, gather_mode
    g0[1] = lds_addr;                                 // LDS dest (bytes)
    g0[2] = (u32)gaddr;                               // global_addr[31:0]
    g0[3] = (u32)((gaddr >> 32) & 0x1FFFFFFu)         // global_addr[56:32]
          | (2u << 30);                               // type = 2 ("image")

    v8i g1;
    g1[0] = (int)(2u << 16);                          // data_size = 4B; wg_mask=0
    g1[1] = (int)((u32)D_DIM << 16);                  // tensor_dim0 = 64   [79:48]
    g1[2] = (int)(((u32)E_ROWS & 0xFFFFu) << 16);     // tensor_dim1 lo16   [111:80]
    g1[3] = (int)(((u32)D_DIM << 16)                  // tile_dim0 = 64     [127:112]
          | ((u32)E_ROWS >> 16));                     // tensor_dim1 hi16
    g1[4] = n_idx;                                    // tile_dim1 = #valid indices
    g1[5] = D_DIM;                                    // tensor_dim0_stride = 64
    g1[6] = 0;
    g1[7] = 0;

    v4i g2 = { i0, i1, i2, i3 };                      // row_index_0..3 (32-bit mode)
    v4i g3 = { i4, i5, i6, i7 };                      // row_index_4..7

    v8i g4 = { 0, 0, 0, 0, 0, 0, 0, 0 };              // VADDR4 slot: unused/NULL

    __builtin_amdgcn_tensor_load_to_lds(g0, g1, g2, g3, g4, /*cpol=*/0);
}

__global__ __launch_bounds__(256) void tbe_pool_fwd_tdm(
    const int*   __restrict__ indices,   // [T*B*L]
    const int*   __restrict__ offsets,   // [T*B + 1]
    const float* __restrict__ weights,   // [T, E, D]
    float*       __restrict__ out)       // [B, T*D]
{
    __shared__ unsigned char smem[WAVES_PER_BLOCK * BAG_BYTES];  // 40 KB

    const int lane = threadIdx.x & 31;
    const int wave = __builtin_amdgcn_readfirstlane((int)(threadIdx.x >> 5));
    const int seg  = (int)blockIdx.x * WAVES_PER_BLOCK + wave;   // bag id
    if (seg >= T_TABLES * B_BATCH) return;

    const int t = seg >> 12;       // seg / 4096
    const int b = seg & 4095;      // seg % 4096

    const int off0 = offsets[seg];
    const int off1 = offsets[seg + 1];
    const int len  = off1 - off0;

    const int*   bag   = indices + off0;
    const float* table = weights + (size_t)t * (size_t)(E_ROWS * D_DIM);

    float accx = 0.0f, accy = 0.0f;

    if (len == BAG_L) {
        // ---- TDM gather path (fixed bag length 20) ----
        int idx[BAG_L];
#pragma unroll
        for (int j = 0; j < BAG_L; ++j) idx[j] = bag[j];          // scalar loads

        const u64 gaddr = (u64)(uintptr_t)table;
        // Flat LDS pointer low 32 bits == wave-relative LDS byte offset.
        const u32 lds0  = (u32)(uintptr_t)(&smem[0]) + (u32)wave * BAG_BYTES;

        // Rows 0..7 -> lds0, 8..15 -> lds0+2048, 16..19 -> lds0+4096
        tdm_gather_rows(gaddr, lds0,            8,
                        idx[0], idx[1], idx[2], idx[3],
                        idx[4], idx[5], idx[6], idx[7]);
        tdm_gather_rows(gaddr, lds0 + 8 * ROW_BYTES, 8,
                        idx[8], idx[9], idx[10], idx[11],
                        idx[12], idx[13], idx[14], idx[15]);
        tdm_gather_rows(gaddr, lds0 + 16 * ROW_BYTES, 4,
                        idx[16], idx[17], idx[18], idx[19],
                        0, 0, 0, 0);

        __builtin_amdgcn_s_wait_tensorcnt(0);

        // ---- Reduce 20 rows from this wave's LDS slice ----
        const float2* rows = (const float2*)(&smem[(size_t)wave * BAG_BYTES]);
        float2 v[BAG_L];
#pragma unroll
        for (int j = 0; j < BAG_L; ++j) v[j] = rows[j * 32 + lane]; // ds_load_b64
#pragma unroll
        for (int j = 0; j < BAG_L; ++j) { accx += v[j].x; accy += v[j].y; }
    } else {
        // ---- Generic fallback: direct global gather, 4-deep pipelined ----
        int j = 0;
        for (; j + 4 <= len; j += 4) {
            const int i0 = bag[j + 0], i1 = bag[j + 1];
            const int i2 = bag[j + 2], i3 = bag[j + 3];
            float2 v0 = ((const float2*)(table + (size_t)i0 * D_DIM))[lane];
            float2 v1 = ((const float2*)(table + (size_t)i1 * D_DIM))[lane];
            float2 v2 = ((const float2*)(table + (size_t)i2 * D_DIM))[lane];
            float2 v3 = ((const float2*)(table + (size_t)i3 * D_DIM))[lane];
            accx += v0.x + v1.x + v2.x + v3.x;
            accy += v0.y + v1.y + v2.y + v3.y;
        }
        for (; j < len; ++j) {
            float2 v = ((const float2*)(table + (size_t)bag[j] * D_DIM))[lane];
            accx += v.x;
            accy += v.y;
        }
    }

    // out[b, t*D + d]; lane covers columns {2*lane, 2*lane+1}.
    float2* o = (float2*)(out + (size_t)b * (T_TABLES * D_DIM) + (size_t)t * D_DIM);
    o[lane] = make_float2(accx, accy);
}

extern "C" void kernel_launch(void* const* d_in, const int* in_sizes, int n_in,
                              void* d_out, int out_size, void* d_ws, size_t ws_size,
                              hipStream_t stream) {
    (void)in_sizes; (void)n_in; (void)out_size; (void)d_ws; (void)ws_size;
    const int*   indices = (const int*)d_in[0];
    const int*   offsets = (const int*)d_in[1];
    const float* weights = (const float*)d_in[2];
    float*       out     = (float*)d_out;

    const int n_segs = T_TABLES * B_BATCH;              // 65536 bags
    const int blocks = n_segs / WAVES_PER_BLOCK;        // 8192 blocks of 256
    tbe_pool_fwd_tdm<<<blocks, 256, 0, stream>>>(indices, offsets, weights, out);
}